// EntropyComputer_14860586844378
// MI455X (gfx1250) — compile-verified
//
#include <hip/hip_runtime.h>
#include <stdint.h>

#define CELLS 81
#define GRIDS_PER_BLOCK 64                 // grid-PAIRS per block
#define THREADS 128
#define HALF_FLOATS (GRIDS_PER_BLOCK * CELLS)      // 5184 floats = 20736 B per input half
#define HALF_CHUNKS (HALF_FLOATS / 4)              // 1296 b128 transfers per half
#define FULL_K (HALF_CHUNKS / THREADS)             // 10 full rounds
#define TAIL_T (HALF_CHUNKS - FULL_K * THREADS)    // 16 tail threads

typedef int v4i __attribute__((vector_size(4 * sizeof(int))));
typedef __attribute__((address_space(1))) v4i GV4;   // global 16B vector
typedef __attribute__((address_space(3))) v4i LV4;   // LDS 16B vector

// ---- CDNA5 async global -> LDS copy (ASYNCcnt path) -------------------------
__device__ __forceinline__ void async_copy_b128(const float* gsrc, float* lds_dst) {
#if __has_builtin(__builtin_amdgcn_global_load_async_to_lds_b128)
    __builtin_amdgcn_global_load_async_to_lds_b128(
        (GV4*)(void*)gsrc, (LV4*)(void*)lds_dst, /*offset=*/0, /*cpol=*/0);
#else
    unsigned lofs = (unsigned)(uintptr_t)(__attribute__((address_space(3))) void*)(void*)lds_dst;
    uint64_t ga   = (uint64_t)(uintptr_t)gsrc;
    asm volatile("global_load_async_to_lds_b128 %0, %1, off"
                 :: "v"(lofs), "v"(ga) : "memory");
#endif
}

__device__ __forceinline__ void wait_async0() {
#if __has_builtin(__builtin_amdgcn_s_wait_asynccnt)
    __builtin_amdgcn_s_wait_asynccnt(0);
#else
    asm volatile("s_wait_asynccnt 0" ::: "memory");
#endif
}

// Branch-free staging of one 20736-byte input half into LDS.
__device__ __forceinline__ void stage_half(const float* __restrict__ src,
                                           float* lds_base) {
    const int tid = threadIdx.x;
#pragma unroll
    for (int k = 0; k < FULL_K; ++k) {
        const int c = tid + k * THREADS;               // chunk index
        async_copy_b128(src + c * 4, lds_base + c * 4);
    }
    if (tid < TAIL_T) {
        const int c = FULL_K * THREADS + tid;
        async_copy_b128(src + c * 4, lds_base + c * 4);
    }
}

// Guarded variant for a (possible) partial tail block.
__device__ __forceinline__ void stage_half_guarded(const float* __restrict__ src,
                                                   float* lds_base,
                                                   long long floats_avail) {
    const int tid = threadIdx.x;
    for (int c = tid; c < HALF_CHUNKS; c += THREADS) {
        if ((long long)(c + 1) * 4 <= floats_avail)
            async_copy_b128(src + c * 4, lds_base + c * 4);
    }
}

// ---- per-grid Sudoku entropy (reads the thread's grid from LDS) -------------
__device__ __forceinline__ float grid_entropy(const float* g) {
    unsigned row[9], col[9], box[9], bits[81];
#pragma unroll
    for (int i = 0; i < 9; ++i) { row[i] = 0u; col[i] = 0u; box[i] = 0u; }

#pragma unroll
    for (int r = 0; r < 9; ++r) {
#pragma unroll
        for (int c = 0; c < 9; ++c) {
            const int v = (int)g[r * 9 + c];        // 0..9 exact in fp32
            const unsigned b = (1u << v) >> 1;      // v==0 -> 0, else bit (v-1)
            bits[r * 9 + c] = b;
            row[r] |= b;
            col[c] |= b;
            box[(r / 3) * 3 + (c / 3)] |= b;
        }
    }

    float ent = 0.0f;
#pragma unroll
    for (int r = 0; r < 9; ++r) {
#pragma unroll
        for (int c = 0; c < 9; ++c) {
            // (row|box) hoists across each 3-column group
            const unsigned forb = (row[r] | box[(r / 3) * 3 + (c / 3)]) | col[c];
            const int n = __popc(~forb & 0x1FFu);   // candidates 0..9
            const float l = __log2f((float)(n > 1 ? n : 1));   // v_log_f32 (TRANS)
            ent += (bits[r * 9 + c] == 0u) ? l : 0.0f;          // empty cell only
        }
    }
    return ent;
}

// ---- kernel -----------------------------------------------------------------
__global__ __launch_bounds__(THREADS) void sudoku_igain_kernel(
    const float* __restrict__ before, const float* __restrict__ after,
    float* __restrict__ out, int batch, long long total_floats)
{
    __shared__ __align__(16) float tile[2 * HALF_FLOATS];   // [before | after]
    __shared__ float hbuf[THREADS];

    const int tid = threadIdx.x;
    const long long base = (long long)blockIdx.x * HALF_FLOATS;
    const long long avail = total_floats - base;

    // ---- stage BOTH input halves up front (both HBM streams in flight) ----
    if (avail >= (long long)HALF_FLOATS) {
        stage_half(before + base, tile);
        stage_half(after + base, tile + HALF_FLOATS);
    } else {
        stage_half_guarded(before + base, tile, avail);
        stage_half_guarded(after + base, tile + HALF_FLOATS, avail);
    }
    wait_async0();
    __syncthreads();

    // ---- threads 0..63 -> before-grids, 64..127 -> after-grids ----
    const int slot = (tid < GRIDS_PER_BLOCK) ? tid
                                             : (HALF_FLOATS / CELLS * CELLS) / CELLS
                                               + (tid - GRIDS_PER_BLOCK); // 64 + t
    const float* g = &tile[(tid < GRIDS_PER_BLOCK ? tid * CELLS
                                                  : HALF_FLOATS + (tid - GRIDS_PER_BLOCK) * CELLS)];
    (void)slot;
    hbuf[tid] = grid_entropy(g);
    __syncthreads();

    if (tid < GRIDS_PER_BLOCK) {
        const int gidx = blockIdx.x * GRIDS_PER_BLOCK + tid;
        if (gidx < batch)
            out[gidx] = hbuf[tid] - hbuf[tid + GRIDS_PER_BLOCK];
    }
}

extern "C" void kernel_launch(void* const* d_in, const int* in_sizes, int n_in,
                              void* d_out, int out_size, void* d_ws, size_t ws_size,
                              hipStream_t stream) {
    (void)n_in; (void)d_ws; (void)ws_size; (void)out_size;
    const float* before = (const float*)d_in[0];
    const float* after  = (const float*)d_in[1];
    float* out = (float*)d_out;

    const int batch = in_sizes[0] / CELLS;                    // 262144
    const long long total_floats = (long long)in_sizes[0];    // batch*81
    const int nblocks = (batch + GRIDS_PER_BLOCK - 1) / GRIDS_PER_BLOCK;  // 4096

    sudoku_igain_kernel<<<nblocks, THREADS, 0, stream>>>(before, after, out,
                                                         batch, total_floats);
}